// GraphAttentionLayer1_11501922418879
// MI455X (gfx1250) — compile-verified
//
#include <hip/hip_runtime.h>
#include <math.h>
#include <stdint.h>

typedef __attribute__((ext_vector_type(2))) float v2f;
typedef __attribute__((ext_vector_type(8))) float v8f;
typedef __attribute__((ext_vector_type(4))) unsigned int v4u;
typedef __attribute__((ext_vector_type(8))) int v8i;
typedef __attribute__((ext_vector_type(4))) int v4i;

constexpr int BB   = 8;     // batch
constexpr int NN   = 2048;  // nodes
constexpr int FIN  = 256;
constexpr int FOUT = 64;
constexpr int TM   = 16;    // row tile
constexpr int CH   = 128;   // j-chunk in attention kernel

#define ALPHA_SLOPE 0.2f

__device__ __forceinline__ float lrelu(float x) { return x > 0.f ? x : ALPHA_SLOPE * x; }

// ---------------------------------------------------------------------------
// Tensor Data Mover path (gfx1250). Guarded: falls back to float4 staging if
// the builtin is absent (also keeps the host pass happy).
// ---------------------------------------------------------------------------
#if defined(__AMDGCN__) && __has_builtin(__builtin_amdgcn_tensor_load_to_lds) && \
    __has_builtin(__builtin_amdgcn_s_wait_tensorcnt)
#define HAVE_TDM 1
#else
#define HAVE_TDM 0
#endif

#if HAVE_TDM
// 1-D contiguous tile: tensor_len / n_elems in 4-byte units.
__device__ __forceinline__ void tdm_load_1d(uint32_t lds_byte_addr,
                                            const float* gsrc,
                                            uint32_t n_elems,
                                            uint32_t tensor_len) {
    const uint64_t ga = (uint64_t)(uintptr_t)gsrc;
    v4u g0;
    g0[0] = 1u;                                            // count=1, user D#, no gather
    g0[1] = lds_byte_addr;                                 // LDS byte address
    g0[2] = (uint32_t)(ga & 0xFFFFFFFFu);                  // global_addr[31:0]
    g0[3] = (uint32_t)((ga >> 32) & 0x01FFFFFFu) | (2u << 30); // addr[56:32] | type=2
    v8i g1;
    g1[0] = (int)(2u << 16);                               // data_size=4B; no mcast/pad
    g1[1] = (int)((tensor_len & 0xFFFFu) << 16);           // tensor_dim0[15:0]
    g1[2] = (int)((tensor_len >> 16) & 0xFFFFu);           // tensor_dim0[31:16]; dim1=0
    g1[3] = (int)((n_elems & 0xFFFFu) << 16);              // tile_dim0
    g1[4] = 0;                                             // tile_dim1/2 = 0 (1-D)
    g1[5] = (int)tensor_len;                               // tensor_dim0_stride[31:0]
    g1[6] = 0;
    g1[7] = 0;
    v4i z4 = {0, 0, 0, 0};
#if __clang_major__ >= 23
    v8i z8 = {0, 0, 0, 0, 0, 0, 0, 0};
    __builtin_amdgcn_tensor_load_to_lds(g0, g1, z4, z4, z8, 0);
#else
    __builtin_amdgcn_tensor_load_to_lds(g0, g1, z4, z4, 0);
#endif
}
#endif

// ---------------------------------------------------------------------------
// Kernel 1: Wh = h @ W^T (16 rows x 64 cols per block) + s1 = Wh.a1, s2 = Wh.a2.
// 4 waves, each a 16x16 slice; K=256 via v_wmma_f32_16x16x4_f32.
// ---------------------------------------------------------------------------
__global__ __launch_bounds__(128) void wh_kernel(
    const float* __restrict__ h, const float* __restrict__ W,
    const float* __restrict__ a, float* __restrict__ Wh,
    float* __restrict__ s1, float* __restrict__ s2)
{
    __shared__ float sh[TM * FIN];     // 16 KB contiguous h tile
    __shared__ float s1t[TM], s2t[TM];

    const int blk  = blockIdx.x;
    const int b    = blk / (NN / TM);
    const int i0   = (blk % (NN / TM)) * TM;
    const int tid  = threadIdx.x;
    const int wave = tid >> 5;
    const int lane = tid & 31;

    const float4* hsrc = reinterpret_cast<const float4*>(h + (size_t)(b * NN + i0) * FIN);
    float4* shv = reinterpret_cast<float4*>(sh);
#pragma unroll
    for (int v = 0; v < 8; ++v) shv[tid + v * 128] = hsrc[tid + v * 128];

    if (tid < TM) { s1t[tid] = 0.f; s2t[tid] = 0.f; }
    __syncthreads();

    const int o0    = wave * 16;
    const int m     = lane & 15;
    const int khalf = (lane >> 4) << 1;     // 0 or 2
    v8f acc = {};

#pragma unroll 4
    for (int k0 = 0; k0 < FIN; k0 += 4) {
        const int kk = k0 + khalf;
        v2f A, Bm;
        A.x  = sh[m * FIN + kk];
        A.y  = sh[m * FIN + kk + 1];
        Bm.x = W[(o0 + m) * FIN + kk];
        Bm.y = W[(o0 + m) * FIN + kk + 1];
        acc = __builtin_amdgcn_wmma_f32_16x16x4_f32(
            false, A, false, Bm, (short)0, acc, false, false);
    }

    const float a1v  = a[o0 + m];
    const float a2v  = a[FOUT + o0 + m];
    const int  rbase = (lane >> 4) << 3;
#pragma unroll
    for (int v = 0; v < 8; ++v) {
        const int row = v + rbase;
        Wh[(size_t)(b * NN + i0 + row) * FOUT + o0 + m] = acc[v];
        float p1 = acc[v] * a1v;
        float p2 = acc[v] * a2v;
#pragma unroll
        for (int off = 8; off >= 1; off >>= 1) {
            p1 += __shfl_xor(p1, off, 16);
            p2 += __shfl_xor(p2, off, 16);
        }
        if (m == 0) { atomicAdd(&s1t[row], p1); atomicAdd(&s2t[row], p2); }
    }
    __syncthreads();
    if (tid < TM) {
        s1[b * NN + i0 + tid] = s1t[tid];
        s2[b * NN + i0 + tid] = s2t[tid];
    }
}

// ---------------------------------------------------------------------------
// Kernel 2: masked-softmax attention + aggregation + ELU for one 16-row tile.
// ---------------------------------------------------------------------------
__global__ __launch_bounds__(256) void attn_kernel(
    const float* __restrict__ Wh, const float* __restrict__ s1g,
    const float* __restrict__ s2g, const int* __restrict__ adj,
    float* __restrict__ out)
{
    __shared__ float s2sh[NN];            // 8 KB
    __shared__ float Whch[CH * FOUT];     // 32 KB : Wh chunk (TDM target)
    __shared__ float Pch[TM * CH];        // 8 KB  : normalized attention chunk
    __shared__ float accsh[TM * FOUT];    // 4 KB
    __shared__ float rowM[TM], rowInvS[TM], s1sh[TM];

    const int blk  = blockIdx.x;
    const int b    = blk / (NN / TM);
    const int i0   = (blk % (NN / TM)) * TM;
    const int tid  = threadIdx.x;
    const int wave = tid >> 5;
    const int lane = tid & 31;

    for (int f = tid; f < NN; f += 256) s2sh[f] = s2g[b * NN + f];
    if (tid < TM) s1sh[tid] = s1g[b * NN + i0 + tid];
    __syncthreads();

    // ---- pass 1: per-row online (max, sum) over masked leakyrelu logits ----
    const int   pi1 = tid >> 4;
    const int   jt  = tid & 15;
    const float s1v = s1sh[pi1];
    float mx = -3.0e38f, sm = 0.f;
    const int4* arow = reinterpret_cast<const int4*>(adj + (size_t)(b * NN + i0 + pi1) * NN);
    for (int it = 0; it < NN / 64; ++it) {
        const int  q4 = jt + it * 16;
        const int4 ad = arow[q4];
        const int  jb = q4 * 4;
        const int adv[4] = {ad.x, ad.y, ad.z, ad.w};
#pragma unroll
        for (int q = 0; q < 4; ++q) {
            if (adv[q] > 0) {
                const float e = lrelu(s1v + s2sh[jb + q]);
                if (e > mx) { sm = sm * __expf(mx - e) + 1.f; mx = e; }
                else        { sm += __expf(e - mx); }
            }
        }
    }
#pragma unroll
    for (int off = 8; off >= 1; off >>= 1) {
        const float om = __shfl_xor(mx, off, 16);
        const float os = __shfl_xor(sm, off, 16);
        const float nm = fmaxf(mx, om);
        sm = sm * __expf(mx - nm) + os * __expf(om - nm);
        mx = nm;
    }
    if (jt == 0) { rowM[pi1] = mx; rowInvS[pi1] = (sm > 0.f) ? 1.f / sm : 0.f; }
    __syncthreads();

    // ---- pass 2: h' = P @ Wh, chunked over j ----
    const int o0    = (wave & 3) * 16;
    const int par   = wave >> 2;
    const int m     = lane & 15;
    const int khalf = (lane >> 4) << 1;
    const int rbase = (lane >> 4) << 3;
    v8f acc = {};

#if HAVE_TDM
    const uint32_t whchLds = (uint32_t)(uintptr_t)&Whch[0];  // flat low 32b = LDS offset
#endif

    for (int c = 0; c < NN / CH; ++c) {
        // prefetch next chunk's adjacency strip (global_prefetch_b8)
        if (c + 1 < NN / CH) {
            const int* nadj = adj + (size_t)(b * NN + i0 + (tid >> 4)) * NN
                              + (c + 1) * CH + (tid & 15) * 8;
            __builtin_prefetch(nadj, 0, 1);
        }

        // stage Wh[c*CH .. +CH][0..63] (contiguous 32 KB) into LDS
#if HAVE_TDM
        if (wave == 0)
            tdm_load_1d(whchLds, Wh + (size_t)(b * NN + c * CH) * FOUT,
                        CH * FOUT, NN * FOUT);
#else
        {
            const float4* wsrc = reinterpret_cast<const float4*>(
                Wh + (size_t)(b * NN + c * CH) * FOUT);
            float4* wdst = reinterpret_cast<float4*>(Whch);
#pragma unroll
            for (int v = 0; v < 8; ++v) wdst[tid + v * 256] = wsrc[tid + v * 256];
        }
#endif

        // compute normalized P chunk (16 x 128) while the TDM runs
#pragma unroll
        for (int v = 0; v < 2; ++v) {
            const int  idx4 = tid + v * 256;
            const int  pi   = idx4 >> 5;
            const int  jc   = (idx4 & 31) * 4;
            const int4 ad   = *reinterpret_cast<const int4*>(
                adj + (size_t)(b * NN + i0 + pi) * NN + c * CH + jc);
            const float base1 = s1sh[pi];
            const float rm = rowM[pi], ri = rowInvS[pi];
            const int adv[4] = {ad.x, ad.y, ad.z, ad.w};
#pragma unroll
            for (int q = 0; q < 4; ++q) {
                const int j = c * CH + jc + q;
                float p = 0.f;
                if (adv[q] > 0) p = __expf(lrelu(base1 + s2sh[j]) - rm) * ri;
                Pch[pi * CH + jc + q] = p;
            }
        }

#if HAVE_TDM
        if (wave == 0) __builtin_amdgcn_s_wait_tensorcnt((short)0);
#endif
        __syncthreads();

        if ((c & 1) == par) {
#pragma unroll 8
            for (int k0 = 0; k0 < CH; k0 += 4) {
                const int kk = k0 + khalf;
                v2f A, Bm;
                A.x  = Pch[m * CH + kk];
                A.y  = Pch[m * CH + kk + 1];
                Bm.x = Whch[kk * FOUT + o0 + m];
                Bm.y = Whch[(kk + 1) * FOUT + o0 + m];
                acc = __builtin_amdgcn_wmma_f32_16x16x4_f32(
                    false, A, false, Bm, (short)0, acc, false, false);
            }
        }
        __syncthreads();
    }

    // merge the two parity waves per output slice, apply ELU, store
    if (wave >= 4) {
#pragma unroll
        for (int v = 0; v < 8; ++v)
            accsh[(v + rbase) * FOUT + o0 + m] = acc[v];
    }
    __syncthreads();
    if (wave < 4) {
#pragma unroll
        for (int v = 0; v < 8; ++v) {
            const int row = v + rbase;
            float r = acc[v] + accsh[row * FOUT + o0 + m];
            r = (r > 0.f) ? r : (__expf(r) - 1.f);   // ELU
            out[(size_t)(b * NN + i0 + row) * FOUT + o0 + m] = r;
        }
    }
}

// ---------------------------------------------------------------------------
extern "C" void kernel_launch(void* const* d_in, const int* in_sizes, int n_in,
                              void* d_out, int out_size, void* d_ws, size_t ws_size,
                              hipStream_t stream) {
    (void)in_sizes; (void)n_in; (void)out_size; (void)ws_size;
    const float* h   = (const float*)d_in[0];
    const float* W   = (const float*)d_in[1];
    const float* a   = (const float*)d_in[2];
    const int*   adj = (const int*)d_in[3];
    float*       out = (float*)d_out;

    float* Wh = (float*)d_ws;                          // B*N*FOUT floats (4 MB)
    float* s1 = Wh + (size_t)BB * NN * FOUT;           // B*N
    float* s2 = s1 + (size_t)BB * NN;                  // B*N

    const dim3 grid(BB * (NN / TM));                   // 1024 tiles
    wh_kernel<<<grid, 128, 0, stream>>>(h, W, a, Wh, s1, s2);
    attn_kernel<<<grid, 256, 0, stream>>>(Wh, s1, s2, adj, out);
}